// SchNetModel_13675175870817
// MI455X (gfx1250) — compile-verified
//
#include <hip/hip_runtime.h>

typedef __attribute__((ext_vector_type(2))) float v2f;
typedef __attribute__((ext_vector_type(8))) float v8f;

#define Bsz 64
#define Asz 512
#define Hsz 128
#define Esz 16384
#define NTILES (Esz / 16)        // 1024 edge tiles
#define NWAVES (Bsz * NTILES)    // 65536 (b, tile) wave tasks

// workspace layout (float offsets)
#define WS_CNT  0                        // 512 floats
#define WS_CARR 512                      // 16384 floats
#define WS_SC   (512 + 16384)            // 1 float
#define WS_V    17408                    // E*H floats (aligned)
#define WS_PART (17408 + Esz * Hsz)      // 65536 floats

// ---- K0: zero the degree counters -----------------------------------------
__global__ void zero_cnt_kernel(float* __restrict__ cnt) {
    int i = blockIdx.x * blockDim.x + threadIdx.x;
    if (i < Asz) cnt[i] = 0.0f;
}

// ---- K1: cnt[a] = #edges with dst==a (exact: small-int f32 atomics) -------
__global__ void count_kernel(const int* __restrict__ dst, float* __restrict__ cnt) {
    int e = blockIdx.x * blockDim.x + threadIdx.x;
    atomicAdd(&cnt[dst[e]], 1.0f);
}

// ---- K2: v[e] = W2 @ (s*wfc_row(dst[e])),  c[e] = s * (b2 . wfc_row) ------
__global__ __launch_bounds__(128) void prep_v_kernel(
    const int* __restrict__ dst, const float* __restrict__ cnt,
    const float* __restrict__ w2, const float* __restrict__ b2,
    const float* __restrict__ wfc, float* __restrict__ vmat,
    float* __restrict__ carr)
{
    __shared__ float uf[Hsz];
    const int e = blockIdx.x;
    const int k = threadIdx.x;
    const int d = dst[e];
    const float s = 1.0f / fmaxf(cnt[d], 1.0f);
    uf[k] = wfc[d * Hsz + k];
    __syncthreads();
    const float* w2row = w2 + k * Hsz;
    float sum = 0.0f;
#pragma unroll 8
    for (int h = 0; h < Hsz; ++h) sum += w2row[h] * uf[h];
    vmat[(size_t)e * Hsz + k] = s * sum;
    if (k == 0) {
        float cs = 0.0f;
        for (int h = 0; h < Hsz; ++h) cs += b2[h] * uf[h];
        carr[e] = s * cs;
    }
}

// ---- K3: Sc = sum_e c[e] ---------------------------------------------------
__global__ __launch_bounds__(256) void reduce_c_kernel(
    const float* __restrict__ carr, float* __restrict__ Sc)
{
    __shared__ float red[256];
    const int tid = threadIdx.x;
    float s = 0.0f;
    for (int i = tid; i < Esz; i += 256) s += carr[i];
    red[tid] = s;
    __syncthreads();
    for (int st = 128; st; st >>= 1) {
        if (tid < st) red[tid] += red[tid + st];
        __syncthreads();
    }
    if (tid == 0) Sc[0] = red[0];
}

// ---- K4: fused gather + (X16x128 @ W1) WMMA + bias + ReLU + dot(v) --------
__global__ __launch_bounds__(256) void fused_msg_kernel(
    const float* __restrict__ x, const int* __restrict__ src,
    const float* __restrict__ w1, const float* __restrict__ b1,
    const float* __restrict__ vmat, float* __restrict__ partials)
{
    // W1 packed in LDS so each B-fragment (rows k, k+1 at col n) is one b64:
    //   w1lds[(k>>1)*256 + 2*n + (k&1)] = w1[k*H + n]
    __shared__ float w1lds[Hsz * Hsz];
    const int tid = threadIdx.x;
    for (int i = tid; i < Hsz * Hsz; i += 256) {
        const int k = i >> 7, n = i & 127;
        w1lds[((k >> 1) << 8) + 2 * n + (k & 1)] = w1[i];
    }
    __syncthreads();

    const int wave = tid >> 5;
    const int lane = tid & 31;
    const int hi   = lane >> 4;   // half-wave select (K pair 0/1 vs 2/3)
    const int lo   = lane & 15;

    const int gtile = blockIdx.x * 8 + wave;   // 0..65535, always in range
    const int b = gtile >> 10;
    const int t = gtile & (NTILES - 1);

    // A-matrix (16x4 f32) fragment source: lane lo -> row M=lo of the tile,
    // VGPR pair holds K = {2*hi, 2*hi+1} of each K4 step.
    const int   erow = t * 16 + lo;
    const int   sidx = src[erow];
    const float* xrow = x + ((size_t)b * Asz + sidx) * Hsz + 2 * hi;

    v2f afr[32];
#pragma unroll
    for (int kk = 0; kk < 32; ++kk)
        afr[kk] = *(const v2f*)(xrow + 4 * kk);

    float dotacc = 0.0f;
    for (int nt = 0; nt < 8; ++nt) {
        const int nbase = nt * 16;
        const float* bbase = &w1lds[hi * 256 + 2 * (nbase + lo)];
        v8f acc = {0.f, 0.f, 0.f, 0.f, 0.f, 0.f, 0.f, 0.f};
#pragma unroll
        for (int kk = 0; kk < 32; ++kk) {
            v2f bfr = *(const v2f*)(bbase + kk * 512);
            acc = __builtin_amdgcn_wmma_f32_16x16x4_f32(
                false, afr[kk], false, bfr, (short)0, acc, false, false);
        }
        // C/D layout: VGPR r, lane L -> M = r + 8*hi, N = lo
        const float  bias = b1[nbase + lo];
        const float* vcol = vmat + (size_t)(t * 16 + 8 * hi) * Hsz + nbase + lo;
#pragma unroll
        for (int r = 0; r < 8; ++r) {
            const float m1 = fmaxf(acc[r] + bias, 0.0f);
            dotacc += m1 * vcol[(size_t)r * Hsz];
        }
    }
#pragma unroll
    for (int off = 16; off; off >>= 1)
        dotacc += __shfl_xor(dotacc, off, 32);
    if (lane == 0) partials[gtile] = dotacc;
}

// ---- K5: out[b] = bfc + Sc + sum of 1024 partials --------------------------
__global__ __launch_bounds__(256) void final_reduce_kernel(
    const float* __restrict__ partials, const float* __restrict__ Sc,
    const float* __restrict__ bfc, float* __restrict__ out)
{
    __shared__ float red[256];
    const int b = blockIdx.x;
    const int tid = threadIdx.x;
    float s = 0.0f;
#pragma unroll
    for (int j = 0; j < 4; ++j) s += partials[b * NTILES + j * 256 + tid];
    red[tid] = s;
    __syncthreads();
    for (int st = 128; st; st >>= 1) {
        if (tid < st) red[tid] += red[tid + st];
        __syncthreads();
    }
    if (tid == 0) out[b] = red[0] + Sc[0] + bfc[0];
}

extern "C" void kernel_launch(void* const* d_in, const int* in_sizes, int n_in,
                              void* d_out, int out_size, void* d_ws, size_t ws_size,
                              hipStream_t stream) {
    const float* x   = (const float*)d_in[0];
    // d_in[1] = pos: unused by the reference output
    const int*   ei  = (const int*)d_in[2];   // (2, E) int32 (JAX default x64 off)
    const float* w1  = (const float*)d_in[3];
    const float* b1  = (const float*)d_in[4];
    const float* w2  = (const float*)d_in[5];
    const float* b2  = (const float*)d_in[6];
    const float* wfc = (const float*)d_in[7];
    const float* bfc = (const float*)d_in[8];
    float* out = (float*)d_out;
    float* ws  = (float*)d_ws;

    const int* srcIdx = ei;
    const int* dstIdx = ei + Esz;

    float* cnt      = ws + WS_CNT;
    float* carr     = ws + WS_CARR;
    float* Sc       = ws + WS_SC;
    float* vmat     = ws + WS_V;
    float* partials = ws + WS_PART;

    zero_cnt_kernel<<<2, 256, 0, stream>>>(cnt);
    count_kernel<<<Esz / 256, 256, 0, stream>>>(dstIdx, cnt);
    prep_v_kernel<<<Esz, 128, 0, stream>>>(dstIdx, cnt, w2, b2, wfc, vmat, carr);
    reduce_c_kernel<<<1, 256, 0, stream>>>(carr, Sc);
    fused_msg_kernel<<<NWAVES / 8, 256, 0, stream>>>(x, srcIdx, w1, b1, vmat, partials);
    final_reduce_kernel<<<Bsz, 256, 0, stream>>>(partials, Sc, bfc, out);
}